// CompAttnSenseNet_30090540876286
// MI455X (gfx1250) — compile-verified
//
#include <hip/hip_runtime.h>
#include <hip/hip_bf16.h>
#include <math.h>

// ---------------------------------------------------------------------------
// CompAttnSenseNet for MI455X (gfx1250, wave32)
//   K1: per-batch attention pipeline -> hidden[256][128]   (L2-resident gathers)
//   K2: logits = hidden @ W_out + b_out via V_WMMA_F32_16X16X4_F32 (branchless)
//   K3: per-row online (max, sumexp) -> logZ
//   K4: out = logits - logZ   (log_softmax, float4-vectorized)
// ---------------------------------------------------------------------------

#define N_L   200
#define N_S   5
#define N_D   128
#define N_LS  (N_L * N_S)
#define N_O   50000
#define PAD_ID 0

typedef float v2f __attribute__((ext_vector_type(2)));
typedef float v8f __attribute__((ext_vector_type(8)));

__device__ __forceinline__ float wave_allreduce_sum(float v) {
    // wave32 butterfly: every lane ends with the full sum
    #pragma unroll
    for (int off = 16; off > 0; off >>= 1)
        v += __shfl_xor(v, off, 32);
    return v;
}

// ---------------------------------------------------------------------------
// Kernel 1: attention pipeline. One workgroup (256 thr = 8 waves) per batch b.
// Lane mapping inside a wave: lane owns dims {lane, lane+32, lane+64, lane+96}.
// ---------------------------------------------------------------------------
__global__ __launch_bounds__(256) void attn_kernel(
    const int* __restrict__ inputs,          // [mb][1000] token ids
    const float* __restrict__ length_w,      // [mb]
    const unsigned char* __restrict__ wmask, // [mb][200] bool
    const float* __restrict__ emb,           // [V][128]
    const float* __restrict__ w_attn,        // [128]
    const float* __restrict__ b_attn_p,      // [1]
    float* __restrict__ hidden)              // [mb][128]
{
    __shared__ float s_gmean[N_D];
    __shared__ float s_ctx[N_D];
    __shared__ float s_imp[256];     // word_imp, then word_w
    __shared__ float s_red[256];
    __shared__ float s_part[8][N_D];

    const int b    = blockIdx.x;
    const int tid  = threadIdx.x;
    const int wave = tid >> 5;
    const int lane = tid & 31;
    const float lw    = length_w[b];
    const float battn = b_attn_p[0];
    const int* tok = inputs + (long)b * N_LS;

    // ---- Phase 1: gmean[d] = lw/S * sum_p emb[tok[p]][d] -------------------
    {
        const int d    = tid & (N_D - 1);
        const int half = tid >> 7;            // 0 or 1
        float acc = 0.f;
        for (int p = half; p < N_LS; p += 2)
            acc += emb[(long)tok[p] * N_D + d];
        s_red[tid] = acc;
        __syncthreads();
        if (tid < N_D)
            s_gmean[tid] = (s_red[tid] + s_red[tid + N_D]) * lw * (1.0f / N_S);
        __syncthreads();
    }

    // ---- Phase 2: word_imp[l] (one wave per word) --------------------------
    for (int l = wave; l < N_L; l += 8) {
        float e[N_S][4];
        float dot[N_S];
        #pragma unroll
        for (int s = 0; s < N_S; ++s) {
            const float* er = emb + (long)tok[l * N_S + s] * N_D;
            float dp = 0.f;
            #pragma unroll
            for (int k = 0; k < 4; ++k) {
                e[s][k] = er[lane + 32 * k];
                dp += e[s][k] * s_gmean[lane + 32 * k];
            }
            dot[s] = wave_allreduce_sum(dp);
        }
        // sense softmax (redundant on all lanes)
        float mx = dot[0];
        #pragma unroll
        for (int s = 1; s < N_S; ++s) mx = fmaxf(mx, dot[s]);
        float w[N_S], norm = 0.f;
        #pragma unroll
        for (int s = 0; s < N_S; ++s) { w[s] = __expf(dot[s] - mx); norm += w[s]; }
        const float inv = 1.0f / norm;
        // word_mean restricted to this lane's dims, dotted with w_attn
        float wi = 0.f;
        #pragma unroll
        for (int k = 0; k < 4; ++k) {
            float wm = 0.f;
            #pragma unroll
            for (int s = 0; s < N_S; ++s) wm += w[s] * inv * e[s][k];
            wi += wm * w_attn[lane + 32 * k];
        }
        wi = wave_allreduce_sum(wi);
        if (lane == 0) s_imp[l] = wi + battn;
    }
    __syncthreads();

    // ---- Phase 3: word softmax over L (with mask) --------------------------
    {
        float v = -INFINITY;
        if (tid < N_L) {
            v = wmask[(long)b * N_L + tid] ? -INFINITY : s_imp[tid];
        }
        s_red[tid] = v;
        __syncthreads();
        for (int off = 128; off > 0; off >>= 1) {
            if (tid < off) s_red[tid] = fmaxf(s_red[tid], s_red[tid + off]);
            __syncthreads();
        }
        const float mx = s_red[0];
        __syncthreads();
        const float ev = (tid < N_L) ? __expf(v - mx) : 0.f;
        s_red[tid] = ev;
        __syncthreads();
        for (int off = 128; off > 0; off >>= 1) {
            if (tid < off) s_red[tid] += s_red[tid + off];
            __syncthreads();
        }
        const float sm = s_red[0];
        __syncthreads();
        if (tid < N_L) s_imp[tid] = ev / sm;   // now word_w
        __syncthreads();
    }

    // ---- Phase 4: context[d] = sum_l word_w[l] * word_mean[l][d] -----------
    {
        float cx[4] = {0.f, 0.f, 0.f, 0.f};
        for (int l = wave; l < N_L; l += 8) {
            float e[N_S][4];
            float dot[N_S];
            #pragma unroll
            for (int s = 0; s < N_S; ++s) {
                const float* er = emb + (long)tok[l * N_S + s] * N_D;
                float dp = 0.f;
                #pragma unroll
                for (int k = 0; k < 4; ++k) {
                    e[s][k] = er[lane + 32 * k];
                    dp += e[s][k] * s_gmean[lane + 32 * k];
                }
                dot[s] = wave_allreduce_sum(dp);
            }
            float mx = dot[0];
            #pragma unroll
            for (int s = 1; s < N_S; ++s) mx = fmaxf(mx, dot[s]);
            float w[N_S], norm = 0.f;
            #pragma unroll
            for (int s = 0; s < N_S; ++s) { w[s] = __expf(dot[s] - mx); norm += w[s]; }
            const float scale = s_imp[l] / norm;   // word_w * sense-softmax norm
            #pragma unroll
            for (int k = 0; k < 4; ++k) {
                float wm = 0.f;
                #pragma unroll
                for (int s = 0; s < N_S; ++s) wm += w[s] * e[s][k];
                cx[k] += scale * wm;
            }
        }
        #pragma unroll
        for (int k = 0; k < 4; ++k) s_part[wave][lane + 32 * k] = cx[k];
        __syncthreads();
        if (tid < N_D) {
            float s = 0.f;
            #pragma unroll
            for (int w8 = 0; w8 < 8; ++w8) s += s_part[w8][tid];
            s_ctx[tid] = s;
        }
        __syncthreads();
    }

    // ---- Phase 5: sense attention vs context -> hidden ---------------------
    {
        float h[4] = {0.f, 0.f, 0.f, 0.f};
        for (int l = wave; l < N_L; l += 8) {
            float e[N_S][4];
            float dot[N_S];
            int   tk[N_S];
            #pragma unroll
            for (int s = 0; s < N_S; ++s) {
                tk[s] = tok[l * N_S + s];
                const float* er = emb + (long)tk[s] * N_D;
                float dp = 0.f;
                #pragma unroll
                for (int k = 0; k < 4; ++k) {
                    e[s][k] = er[lane + 32 * k];
                    dp += e[s][k] * s_ctx[lane + 32 * k];
                }
                dot[s] = wave_allreduce_sum(dp);
            }
            float mx = dot[0];
            #pragma unroll
            for (int s = 1; s < N_S; ++s) mx = fmaxf(mx, dot[s]);
            float w[N_S], norm = 0.f;
            #pragma unroll
            for (int s = 0; s < N_S; ++s) { w[s] = __expf(dot[s] - mx); norm += w[s]; }
            const float inv = lw / norm;
            #pragma unroll
            for (int s = 0; s < N_S; ++s) {
                float a = (tk[s] == PAD_ID) ? 0.f : w[s] * inv;
                #pragma unroll
                for (int k = 0; k < 4; ++k) h[k] += a * e[s][k];
            }
        }
        #pragma unroll
        for (int k = 0; k < 4; ++k) s_part[wave][lane + 32 * k] = h[k];
        __syncthreads();
        if (tid < N_D) {
            float s = 0.f;
            #pragma unroll
            for (int w8 = 0; w8 < 8; ++w8) s += s_part[w8][tid];
            hidden[(long)b * N_D + tid] = s;
        }
    }
}

// ---------------------------------------------------------------------------
// Kernel 2: logits = hidden(256x128) @ W_out(128x50000) + b_out
// FP32 WMMA 16x16x4. Block = 8 waves; wave w owns a 16(M)x16(N) tile, block
// covers 16(M) x 128(N). K=128 -> 32 WMMA steps per wave.
//
// A layout (16x4 f32): lanes 0-15 M=0..15 K={0,1}; lanes 16-31 M=0..15 K={2,3}
// B layout (4x16 f32): vgpr j, lane-half h -> K = j + 2h, N = lane&15
// C/D: vgpr r, lane-half h -> M = r + 8h, N = lane&15
//
// Branchless OOB handling: a WMMA B-column only feeds the same D-column, so
// out-of-range lanes load a *clamped* in-bounds column unconditionally; their
// (garbage) D columns are simply never stored. This keeps EXEC untouched in
// the hot loop (no s_and_saveexec per load pair) and lets loads clause.
// ---------------------------------------------------------------------------
__global__ __launch_bounds__(256) void gemm_wmma_kernel(
    const float* __restrict__ hidden,   // [256][128]
    const float* __restrict__ W,        // [128][50000]
    const float* __restrict__ bias,     // [50000]
    float* __restrict__ logits)         // [256][50000]
{
    const int tid  = threadIdx.x;
    const int wave = tid >> 5;
    const int lane = tid & 31;
    const int h    = lane >> 4;         // lane half
    const int l15  = lane & 15;

    const int m0   = blockIdx.y * 16;
    const int n0   = blockIdx.x * 128 + wave * 16;
    const int col  = n0 + l15;
    const int row  = m0 + l15;
    const bool colok = (col < N_O);
    const int colc   = colok ? col : (N_O - 1);   // clamped, always in-bounds

    const float* __restrict__ Arow = hidden + (long)row * N_D;
    const float* __restrict__ Wcol = W + colc;

    v8f c = {0.f, 0.f, 0.f, 0.f, 0.f, 0.f, 0.f, 0.f};

    #pragma unroll 4
    for (int kk = 0; kk < 32; ++kk) {
        const int kbase = kk * 4 + h * 2;             // even -> 8B-aligned
        const v2f a = *(const v2f*)(Arow + kbase);    // A: K pair {kbase,kbase+1}
        v2f bb;
        bb.x = Wcol[(long)kbase * N_O];               // B row kbase
        bb.y = Wcol[(long)(kbase + 1) * N_O];         // B row kbase+1
        // D = A*B + C, f32 accumulate (matches reference precision)
        c = __builtin_amdgcn_wmma_f32_16x16x4_f32(
                false, a, false, bb, (short)0, c, false, false);
    }

    if (colok) {
        const float bs = bias[col];
        #pragma unroll
        for (int r = 0; r < 8; ++r) {
            const int m = m0 + r + h * 8;
            logits[(long)m * N_O + col] = c[r] + bs;
        }
    }
}

// ---------------------------------------------------------------------------
// Kernel 3: per-row online logsumexp -> rowz[b] = max + log(sum exp(x-max))
// ---------------------------------------------------------------------------
__global__ __launch_bounds__(256) void rowz_kernel(
    const float* __restrict__ logits, float* __restrict__ rowz)
{
    __shared__ float rm[256], rs[256];
    const int b   = blockIdx.x;
    const int tid = threadIdx.x;
    const float* x = logits + (long)b * N_O;

    float m = -INFINITY, s = 0.f;
    for (int i = tid; i < N_O; i += 256) {
        const float v  = x[i];
        const float nm = fmaxf(m, v);
        s = s * __expf(m - nm) + __expf(v - nm);
        m = nm;
    }
    rm[tid] = m; rs[tid] = s;
    __syncthreads();
    for (int off = 128; off > 0; off >>= 1) {
        if (tid < off) {
            const float m2 = rm[tid + off], s2 = rs[tid + off];
            const float nm = fmaxf(rm[tid], m2);
            rs[tid] = rs[tid] * __expf(rm[tid] - nm) + s2 * __expf(m2 - nm);
            rm[tid] = nm;
        }
        __syncthreads();
    }
    if (tid == 0) rowz[b] = rm[0] + __logf(rs[0]);
}

// ---------------------------------------------------------------------------
// Kernel 4: log_softmax in place: out -= rowz[row]  (float4 vectorized;
// N_O % 4 == 0, so a float4 never straddles a row boundary)
// ---------------------------------------------------------------------------
__global__ __launch_bounds__(256) void norm_kernel(
    float4* __restrict__ out, const float* __restrict__ rowz, long total4)
{
    const long i = (long)blockIdx.x * blockDim.x + threadIdx.x;
    if (i < total4) {
        const float z = rowz[(i * 4) / N_O];
        float4 v = out[i];
        v.x -= z; v.y -= z; v.z -= z; v.w -= z;
        out[i] = v;
    }
}

// ---------------------------------------------------------------------------
extern "C" void kernel_launch(void* const* d_in, const int* in_sizes, int n_in,
                              void* d_out, int out_size, void* d_ws, size_t ws_size,
                              hipStream_t stream)
{
    const int*           inputs = (const int*)d_in[0];
    const float*         lw     = (const float*)d_in[1];
    const unsigned char* wmask  = (const unsigned char*)d_in[2];
    const float*         emb    = (const float*)d_in[3];
    const float*         Wout   = (const float*)d_in[4];
    const float*         bout   = (const float*)d_in[5];
    const float*         wattn  = (const float*)d_in[6];
    const float*         battn  = (const float*)d_in[7];
    float*               out    = (float*)d_out;

    const int mb = in_sizes[0] / N_LS;      // 256

    float* hidden = (float*)d_ws;           // mb * 128 floats
    float* rowz   = hidden + (long)mb * N_D;

    attn_kernel<<<mb, 256, 0, stream>>>(inputs, lw, wmask, emb, wattn, battn, hidden);

    dim3 ggrid((N_O + 127) / 128, mb / 16);
    gemm_wmma_kernel<<<ggrid, 256, 0, stream>>>(hidden, Wout, bout, out);

    rowz_kernel<<<mb, 256, 0, stream>>>(out, rowz);

    const long total4 = ((long)mb * N_O) / 4;
    norm_kernel<<<(int)((total4 + 255) / 256), 256, 0, stream>>>(
        (float4*)out, rowz, total4);
}